// proposal_layer_40080634806594
// MI455X (gfx1250) — compile-verified
//
#include <hip/hip_runtime.h>
#include <hip/hip_bf16.h>
#include <stdint.h>

#define PRE_N   6000
#define POST_N  300
#define NMS_TH  0.7f
#define MAX_NW  ((PRE_N + 31) / 32)   // 188 suppression words per row

typedef __attribute__((ext_vector_type(4))) unsigned int u32x4;
typedef __attribute__((ext_vector_type(8))) int          i32x8;
typedef __attribute__((ext_vector_type(4))) int          i32x4;

#if __has_builtin(__builtin_amdgcn_tensor_load_to_lds) && __has_builtin(__builtin_amdgcn_s_wait_tensorcnt)
#define HAVE_TDM 1
#endif

#ifdef HAVE_TDM
// Generic TDM descriptor build + issue: 1D (tile1 == 0) or 2D tile.
// dsz: 0=1B 1=2B 2=4B 3=8B. tensor dims give OOB zero-padding.
static __device__ __forceinline__ void tdm_load(uint32_t lds_off, uint64_t ga,
                                                uint32_t dsz, uint32_t tdim0,
                                                uint32_t tdim1, uint32_t tile0,
                                                uint32_t tile1, uint32_t stride0) {
  u32x4 g0;
  g0.x = 1u;                                            // count=1, user descriptor
  g0.y = lds_off;                                       // lds_addr (bytes)
  g0.z = (uint32_t)ga;                                  // global_addr[31:0]
  g0.w = (uint32_t)((ga >> 32) & 0x01ffffffull)         // global_addr[56:32]
       | (2u << 30);                                    // type = 2 ("image")
  i32x8 g1;
  g1[0] = (int)(dsz << 16);                             // data_size
  g1[1] = (int)((tdim0 & 0xffffu) << 16);               // tensor_dim0 lo16
  g1[2] = (int)((tdim0 >> 16) & 0xffffu)                // tensor_dim0 hi16
        | (int)((tdim1 & 0xffffu) << 16);               // tensor_dim1 lo16
  g1[3] = (int)((tdim1 >> 16) & 0xffffu)                // tensor_dim1 hi16
        | (int)((tile0 & 0xffffu) << 16);               // tile_dim0
  g1[4] = (int)(tile1 & 0xffffu);                       // tile_dim1 (0 = 1D)
  g1[5] = (int)stride0;                                 // tensor_dim0_stride lo32
  g1[6] = 0;
  g1[7] = 0;
  i32x4 gz = (i32x4)0;
#if defined(__clang_major__) && (__clang_major__ >= 23)
  i32x8 gz8 = (i32x8)0;
  __builtin_amdgcn_tensor_load_to_lds(g0, g1, gz, gz, gz8, 0);
#else
  __builtin_amdgcn_tensor_load_to_lds(g0, g1, gz, gz, 0);
#endif
}
#endif

// Monotonic (descending-score, ascending-index tiebreak) 64-bit sort key,
// matching lax.top_k's stable ordering.
static __device__ __forceinline__ uint64_t sort_key(float s, int i) {
  uint32_t u = __float_as_uint(s);
  u = (u & 0x80000000u) ? ~u : (u | 0x80000000u);
  return ((uint64_t)u << 32) | (uint32_t)(~(uint32_t)i);
}

// ---------------- kernel 1: build sort keys from cls_prob[..., 1] ----------
__global__ void k_keys(const float* __restrict__ cls,
                       uint64_t* __restrict__ keys, int A) {
  int i = blockIdx.x * blockDim.x + threadIdx.x;
  if (i < A) keys[i] = sort_key(cls[2 * i + 1], i);
}

// ---------------- kernel 2: exact top-n via rank counting ------------------
// rank(i) = #{j : key_j > key_i}; keys are unique, so ranks are a permutation.
// Key tiles are staged into LDS by the Tensor Data Mover, double-buffered:
// wave 0 issues the DMA for tile t+1 while all waves compare against tile t
// (s_wait_tensorcnt 1 keeps exactly one transfer in flight).
__global__ void __launch_bounds__(256)
k_rank(const uint64_t* __restrict__ keys, int* __restrict__ sortedIdx,
       int A, int topn) {
  __shared__ uint64_t tile[2][256];
  int i = blockIdx.x * blockDim.x + threadIdx.x;
  uint64_t my = (i < A) ? keys[i] : 0ull;
  int rank = 0;
  const int numTiles = (A + 255) >> 8;

#ifdef HAVE_TDM
  if (threadIdx.x < 32)                     // wave 0 drives the DMA
    tdm_load((uint32_t)(uintptr_t)&tile[0][0], (uint64_t)(uintptr_t)keys,
             3u, (uint32_t)A, 1u, 256u, 0u, 256u);
  for (int t = 0; t < numTiles; ++t) {
    const int base = t << 8;
    if (threadIdx.x < 32) {
      if (t + 1 < numTiles) {
        tdm_load((uint32_t)(uintptr_t)&tile[(t + 1) & 1][0],
                 (uint64_t)(uintptr_t)(keys + base + 256),
                 3u, (uint32_t)(A - base - 256), 1u, 256u, 0u, 256u);
        __builtin_amdgcn_s_wait_tensorcnt(1);   // tile t complete (in-order)
      } else {
        __builtin_amdgcn_s_wait_tensorcnt(0);   // last tile complete
      }
    }
    __syncthreads();
    const uint64_t* tt = tile[t & 1];
    const int lim = min(256, A - base);
#pragma unroll 4
    for (int c = 0; c < lim; ++c) rank += (tt[c] > my) ? 1 : 0;
    __syncthreads();
  }
#else
  for (int base = 0; base < A; base += 256) {
    int j = base + (int)threadIdx.x;
    tile[0][threadIdx.x] = (j < A) ? keys[j] : 0ull;
    if (base + 256 + (int)threadIdx.x < A)
      __builtin_prefetch(&keys[base + 256 + threadIdx.x], 0, 0);
    __syncthreads();
    const int lim = min(256, A - base);
#pragma unroll 4
    for (int c = 0; c < lim; ++c) rank += (tile[0][c] > my) ? 1 : 0;
    __syncthreads();
  }
#endif

  if (i < A && rank < topn) sortedIdx[rank] = i;
}

// ---------------- kernel 3: delta2bbox + clip ------------------------------
__global__ void k_decode(const int* __restrict__ sortedIdx,
                         const float* __restrict__ anchors,
                         const float* __restrict__ deltas,
                         const int* __restrict__ img,
                         float* __restrict__ boxes, int n) {
  int r = blockIdx.x * blockDim.x + threadIdx.x;
  if (r >= n) return;
  int i = sortedIdx[r];
  float a0 = anchors[4 * i + 0], a1 = anchors[4 * i + 1];
  float a2 = anchors[4 * i + 2], a3 = anchors[4 * i + 3];
  float dx = deltas[4 * i + 0], dy = deltas[4 * i + 1];
  float dw = deltas[4 * i + 2], dh = deltas[4 * i + 3];
  float w  = a2 - a0 + 1.0f, h = a3 - a1 + 1.0f;
  float cx = a0 + 0.5f * w,  cy = a1 + 0.5f * h;
  float pcx = dx * w + cx,   pcy = dy * h + cy;
  float pw  = expf(dw) * w,  ph  = expf(dh) * h;
  float x1 = pcx - 0.5f * pw, y1 = pcy - 0.5f * ph;
  float x2 = pcx + 0.5f * pw, y2 = pcy + 0.5f * ph;
  float mx = (float)img[1] - 1.0f, my = (float)img[0] - 1.0f;
  x1 = fminf(fmaxf(x1, 0.f), mx);  y1 = fminf(fmaxf(y1, 0.f), my);
  x2 = fminf(fmaxf(x2, 0.f), mx);  y2 = fminf(fmaxf(y2, 0.f), my);
  boxes[4 * r + 0] = x1; boxes[4 * r + 1] = y1;
  boxes[4 * r + 2] = x2; boxes[4 * r + 3] = y2;
}

// ---------------- kernel 4: 32x32-tile suppression bitmask -----------------
// One wave32 per tile: thread t owns row rowBlock*32+t and emits a 32-bit
// word of (iou > thresh && col > row) bits. Column boxes are staged into LDS
// by TDM (2D descriptor: 4B elements, 4-wide rows, 32 rows;
// tensor_dim1 = remaining rows so TDM OOB semantics zero-pad the last tile).
__global__ void __launch_bounds__(32)
k_mask(const float* __restrict__ boxes, uint32_t* __restrict__ mask,
       int n, int nw) {
  __shared__ float colb[32 * 4];
  const int colBlock = blockIdx.x, rowBlock = blockIdx.y;
  const int col0 = colBlock * 32;

#ifdef HAVE_TDM
  {
    const int remain = n - col0;                   // >= 1 (grid sized by nw)
    tdm_load((uint32_t)(uintptr_t)&colb[0],
             (uint64_t)(uintptr_t)(boxes + (size_t)col0 * 4),
             2u, 4u, (uint32_t)remain, 4u, 32u, 4u);
    __builtin_amdgcn_s_wait_tensorcnt(0);
  }
  __syncthreads();
#else
  {
    int c = (int)threadIdx.x, col = col0 + c;
    float4 b = make_float4(0.f, 0.f, 0.f, 0.f);
    if (col < n) b = *(const float4*)(boxes + 4 * col);
    colb[4 * c + 0] = b.x; colb[4 * c + 1] = b.y;
    colb[4 * c + 2] = b.z; colb[4 * c + 3] = b.w;
  }
  __syncthreads();
#endif

  int row = rowBlock * 32 + (int)threadIdx.x;
  float r0 = 0.f, r1 = 0.f, r2 = 0.f, r3 = 0.f, areaR = 0.f;
  if (row < n) {
    r0 = boxes[4 * row + 0]; r1 = boxes[4 * row + 1];
    r2 = boxes[4 * row + 2]; r3 = boxes[4 * row + 3];
    areaR = (r2 - r0) * (r3 - r1);
  }
  uint32_t bits = 0u;
#pragma unroll 8
  for (int c = 0; c < 32; ++c) {
    float c0 = colb[4 * c + 0], c1 = colb[4 * c + 1];
    float c2 = colb[4 * c + 2], c3 = colb[4 * c + 3];
    float areaC = (c2 - c0) * (c3 - c1);
    float iw = fmaxf(fminf(r2, c2) - fmaxf(r0, c0), 0.f);
    float ih = fmaxf(fminf(r3, c3) - fmaxf(r1, c1), 0.f);
    float inter = iw * ih;
    float iou = inter / (areaR + areaC - inter + 1e-8f);
    int col = col0 + c;
    if (col > row && col < n && iou > NMS_TH) bits |= (1u << c);
  }
  if (row < n) mask[(size_t)row * nw + colBlock] = bits;
}

// ---------------- kernel 5: sequential scan + emit ROIs --------------------
__global__ void __launch_bounds__(192)
k_scan_out(const uint32_t* __restrict__ mask, const float* __restrict__ boxes,
           float* __restrict__ out, int n, int nw) {
  __shared__ uint32_t remv[MAX_NW];
  __shared__ int list[POST_N];
  __shared__ int sCnt, sKeep, sStop;
  int tid = (int)threadIdx.x;
  if (tid < nw) remv[tid] = 0u;
  if (tid == 0) { sCnt = 0; sStop = 0; }
  __syncthreads();
  for (int i = 0; i < n; ++i) {
    if (tid == 0) {
      sKeep = 0;
      if (!((remv[i >> 5] >> (i & 31)) & 1u)) {
        sKeep = 1;
        list[sCnt++] = i;
        if (sCnt == POST_N) sStop = 1;   // 300th keep: later boxes irrelevant
      }
    }
    __syncthreads();
    if (sStop) break;                    // uniform across block
    if (sKeep && tid < nw) remv[tid] |= mask[(size_t)i * nw + tid];
    __syncthreads();
  }
  __syncthreads();
  int cnt = sCnt;
  for (int k = tid; k < POST_N; k += (int)blockDim.x) {
    float o0 = 0.f, o1 = 0.f, o2 = 0.f, o3 = 0.f;
    if (k < cnt) {
      int b = list[k];
      o0 = boxes[4 * b + 0]; o1 = boxes[4 * b + 1];
      o2 = boxes[4 * b + 2]; o3 = boxes[4 * b + 3];
    }
    out[4 * k + 0] = o0; out[4 * k + 1] = o1;
    out[4 * k + 2] = o2; out[4 * k + 3] = o3;
  }
}

// ---------------- host launcher --------------------------------------------
extern "C" void kernel_launch(void* const* d_in, const int* in_sizes, int n_in,
                              void* d_out, int out_size, void* d_ws, size_t ws_size,
                              hipStream_t stream) {
  const float* cls     = (const float*)d_in[0];  // (1,fh,fw,na,2)
  const float* deltas  = (const float*)d_in[1];  // (1,fh,fw,na,4)
  const float* anchors = (const float*)d_in[2];  // (A,4)
  const int*   img     = (const int*)d_in[3];    // [H, W]
  float* out = (float*)d_out;

  const int A = in_sizes[2] / 4;
  const int n = (PRE_N < A) ? PRE_N : A;
  const int nw = (n + 31) / 32;

  char* ws = (char*)d_ws;
  size_t off = 0;
  uint64_t* keys = (uint64_t*)(ws + off);
  off = (off + (size_t)A * 8 + 255) & ~(size_t)255;
  int* sortedIdx = (int*)(ws + off);
  off = (off + (size_t)n * 4 + 255) & ~(size_t)255;
  float* boxes = (float*)(ws + off);
  off = (off + (size_t)n * 16 + 255) & ~(size_t)255;
  uint32_t* mask = (uint32_t*)(ws + off);          // n * nw words (~4.5 MB)

  k_keys<<<(A + 255) / 256, 256, 0, stream>>>(cls, keys, A);
  k_rank<<<(A + 255) / 256, 256, 0, stream>>>(keys, sortedIdx, A, n);
  k_decode<<<(n + 255) / 256, 256, 0, stream>>>(sortedIdx, anchors, deltas, img, boxes, n);
  dim3 g(nw, (n + 31) / 32);
  k_mask<<<g, 32, 0, stream>>>(boxes, mask, n, nw);
  k_scan_out<<<1, 192, 0, stream>>>(mask, boxes, out, n, nw);
}